// ModelW_35751307772139
// MI455X (gfx1250) — compile-verified
//
#include <hip/hip_runtime.h>
#include <math.h>

typedef __attribute__((ext_vector_type(2))) float v2f;
typedef __attribute__((ext_vector_type(4))) float v4f;
typedef __attribute__((ext_vector_type(8))) float v8f;

#define SQRT5F   2.23606797749979f
#define FIVE3F   1.6666666666666667f          // 5/3  (s^2/3 = (5/3) d^2)
#define NEXP2C  -3.2262686157437555f          // -sqrt(5) * log2(e)
#define NSPLIT   4

// Pass 1: scale inputs by 1/lengthscale (pad to stride 8), row norms,
// domain labels, and folded alpha*T table aT[(ND+1)][N].
__global__ void precompute_kernel(
    const float* __restrict__ x, const float* __restrict__ xobs,
    const float* __restrict__ alpha, const float* __restrict__ tfl,
    const float* __restrict__ ls, const int* __restrict__ dr_obs,
    const int* __restrict__ dr_x,
    int N, int M, int D, int ND,
    float* __restrict__ xobs_s, float* __restrict__ x_s,
    float* __restrict__ n_obs, float* __restrict__ n_x,
    float* __restrict__ aT, int* __restrict__ lab_x)
{
    const int i = blockIdx.x * blockDim.x + threadIdx.x;
    const float inv_ls = 1.0f / ls[0];
    if (i < N) {
        float nrm = 0.0f;
        #pragma unroll
        for (int k = 0; k < 8; ++k) {
            float v = (k < D) ? xobs[(size_t)i * D + k] * inv_ls : 0.0f;
            xobs_s[(size_t)i * 8 + k] = v;
            nrm += v * v;
        }
        n_obs[i] = nrm;
        int lab = ND;
        for (int d = 0; d < ND; ++d)
            if (i >= dr_obs[2 * d] && i < dr_obs[2 * d + 1]) lab = d;
        const float a = alpha[i];
        for (int dj = 0; dj <= ND; ++dj) {
            float w = 1.0f;
            if (dj != lab && dj < ND && lab < ND) {
                int lo = lab < dj ? lab : dj;
                int hi = lab < dj ? dj : lab;
                int idx = lo * ND - (lo * (lo + 1)) / 2 + (hi - lo - 1);
                w = 1.0f / (1.0f + expf(-tfl[idx]));   // sigmoid
            }
            aT[(size_t)dj * N + i] = a * w;
        }
    }
    if (i < M) {
        float nrm = 0.0f;
        #pragma unroll
        for (int k = 0; k < 8; ++k) {
            float v = (k < D) ? x[(size_t)i * D + k] * inv_ls : 0.0f;
            x_s[(size_t)i * 8 + k] = v;
            nrm += v * v;
        }
        n_x[i] = nrm;
        int lab = ND;
        for (int d = 0; d < ND; ++d)
            if (i >= dr_x[2 * d] && i < dr_x[2 * d + 1]) lab = d;
        lab_x[i] = lab;
    }
}

// Pass 2: one wave per (16-column tile, N-chunk). WMMA f32 16x16x4 (x2 for
// D=8) builds the 16x16 tile of dot products; raw TRANS ops (v_sqrt_f32,
// v_exp_f32) finish the Matern-5/2 value and alpha-weighted row reduction.
__global__ void __launch_bounds__(256)
matern_dot_kernel(const float* __restrict__ xobs_s, const float* __restrict__ x_s,
                  const float* __restrict__ n_obs, const float* __restrict__ n_x,
                  const float* __restrict__ aT, const int* __restrict__ lab_x,
                  float* __restrict__ part, int N, int M, int chunkRows)
{
    const int lane = threadIdx.x & 31;
    const int wave = blockIdx.x * (blockDim.x >> 5) + (threadIdx.x >> 5);
    const int colTiles = (M + 15) >> 4;
    const int tile  = wave % colTiles;
    const int split = wave / colTiles;
    if (split >= NSPLIT) return;               // wave-uniform
    const int jb = tile * 16;

    int col = jb + (lane & 15);
    if (col >= M) col = M - 1;                 // clamp (M=8192 -> no-op)
    const int koff = (lane < 16) ? 0 : 2;      // k-half held by this lane
    const int hi8  = (lane < 16) ? 0 : 8;      // C/D row offset for this lane

    // B operand: loop-invariant column block, scaled x
    const float* bp = x_s + (size_t)col * 8 + koff;
    const v2f blo = *(const v2f*)bp;           // k = koff, koff+1
    const v2f bhi = *(const v2f*)(bp + 4);     // k = koff+4, koff+5
    const float nb = n_x[col];
    const float* wrow = aT + (size_t)lab_x[col] * N;

    const int iBeg = split * chunkRows;
    int iEnd = iBeg + chunkRows;
    if (iEnd > N) iEnd = N;

    float partial = 0.0f;
    for (int i0 = iBeg; i0 < iEnd; i0 += 16) {
        const int row = i0 + (lane & 15);
        const float* ap = xobs_s + (size_t)row * 8 + koff;
        const v2f alo = *(const v2f*)ap;
        const v2f ahi = *(const v2f*)(ap + 4);

        v8f acc = {};
        acc = __builtin_amdgcn_wmma_f32_16x16x4_f32(false, alo, false, blo,
                                                    (short)0, acc, false, false);
        acc = __builtin_amdgcn_wmma_f32_16x16x4_f32(false, ahi, false, bhi,
                                                    (short)0, acc, false, false);

        const int rb = i0 + hi8;               // rows rb..rb+7 for this lane half
        const v4f nr0 = *(const v4f*)(n_obs + rb);
        const v4f nr1 = *(const v4f*)(n_obs + rb + 4);
        const v4f w0  = *(const v4f*)(wrow + rb);
        const v4f w1  = *(const v4f*)(wrow + rb + 4);

        #pragma unroll
        for (int v = 0; v < 8; ++v) {
            const float nr = (v < 4) ? nr0[v] : nr1[v - 4];
            const float wa = (v < 4) ? w0[v]  : w1[v - 4];
            float sq = nr + nb - 2.0f * acc[v];
            sq = fmaxf(sq, 1e-12f);
            const float dd = __builtin_amdgcn_sqrtf(sq);          // v_sqrt_f32
            const float ex = __builtin_amdgcn_exp2f(NEXP2C * dd); // v_exp_f32
            // 1 + sqrt5*d + (5/3)*d^2, two FMAs
            const float kv = fmaf(FIVE3F * dd, dd, fmaf(SQRT5F, dd, 1.0f));
            partial = fmaf(wa * kv, ex, partial);
        }
    }

    // lanes L and L+16 hold the two row-halves of the same column
    partial += __shfl_xor(partial, 16, 32);
    if (lane < 16 && (jb + lane) < M)
        part[(size_t)split * M + jb + lane] = partial;
}

// Pass 3: deterministic fixed-order reduction of the NSPLIT partials.
__global__ void reduce_kernel(const float* __restrict__ part,
                              float* __restrict__ out, int M)
{
    const int j = blockIdx.x * blockDim.x + threadIdx.x;
    if (j >= M) return;
    float s = 0.0f;
    #pragma unroll
    for (int c = 0; c < NSPLIT; ++c) s += part[(size_t)c * M + j];
    out[j] = s;
}

extern "C" void kernel_launch(void* const* d_in, const int* in_sizes, int n_in,
                              void* d_out, int out_size, void* d_ws, size_t ws_size,
                              hipStream_t stream) {
    const float* x     = (const float*)d_in[0];
    const float* xobs  = (const float*)d_in[1];
    const float* alpha = (const float*)d_in[2];
    const float* tfl   = (const float*)d_in[3];
    const float* ls    = (const float*)d_in[4];
    const int*   dro   = (const int*)d_in[5];
    const int*   drx   = (const int*)d_in[6];

    const int N  = in_sizes[2];            // alpha is [N,1]
    const int M  = out_size;               // output is [1,M]
    const int D  = in_sizes[1] / N;        // xobs is [N,D]
    const int ND = in_sizes[5] / 2;        // domain_ranges_obs is [ND,2]

    // workspace layout (floats)
    float* ws      = (float*)d_ws;
    float* xobs_s  = ws;                               // N*8
    float* x_s     = xobs_s + (size_t)N * 8;           // M*8
    float* n_obs   = x_s    + (size_t)M * 8;           // N
    float* n_x     = n_obs  + N;                       // M
    float* aT      = n_x    + M;                       // (ND+1)*N
    int*   labx    = (int*)(aT + (size_t)(ND + 1) * N);// M
    float* part    = (float*)(labx + M);               // NSPLIT*M

    const int nmax = N > M ? N : M;
    precompute_kernel<<<(nmax + 255) / 256, 256, 0, stream>>>(
        x, xobs, alpha, tfl, ls, dro, drx, N, M, D, ND,
        xobs_s, x_s, n_obs, n_x, aT, labx);

    // chunk rows: multiple of 16 so every wave's EXEC stays full in the loop
    const int rowTiles  = (N + 15) / 16;
    const int chunkRows = ((rowTiles + NSPLIT - 1) / NSPLIT) * 16;

    const int colTiles = (M + 15) / 16;
    const int waves    = colTiles * NSPLIT;
    const int blocks   = (waves + 7) / 8;  // 8 waves (256 threads) per block
    matern_dot_kernel<<<blocks, 256, 0, stream>>>(
        xobs_s, x_s, n_obs, n_x, aT, labx, part, N, M, chunkRows);

    reduce_kernel<<<(M + 255) / 256, 256, 0, stream>>>(part, (float*)d_out, M);
}